// GNNModel_38199439130939
// MI455X (gfx1250) — compile-verified
//
#include <hip/hip_runtime.h>

// ---------------------------------------------------------------------------
// GCN 2-layer forward for MI455X (gfx1250, wave32).
// GEMMs use V_WMMA_F32_16X16X4_F32 (full fp32 matrix pipe, matches fp32 ref).
// Aggregation uses wave-per-edge coalesced gather + global_atomic_add_f32.
// ---------------------------------------------------------------------------

typedef float v2f __attribute__((ext_vector_type(2)));
typedef float v8f __attribute__((ext_vector_type(8)));

#define N_NODES 50000
#define N_EDGES 800000
#define C_IN    128
#define C_HID   128
#define C_OUTC  64

// ---------------- degree / normalization ----------------

__global__ void zero_kernel(float* __restrict__ p, int n) {
    int i = blockIdx.x * blockDim.x + threadIdx.x;
    if (i < n) p[i] = 0.0f;
}

__global__ void deg_kernel(const int* __restrict__ dst, float* __restrict__ deg, int E) {
    int i = blockIdx.x * blockDim.x + threadIdx.x;
    if (i < E) atomicAdd(&deg[dst[i]], 1.0f);
}

// in-place: deg -> 1/sqrt(deg + 1)   (+1 = self loop; always > 0)
__global__ void invsqrt_kernel(float* __restrict__ deg, int n) {
    int i = blockIdx.x * blockDim.x + threadIdx.x;
    if (i < n) deg[i] = rsqrtf(deg[i] + 1.0f);
}

// ---------------- fp32 WMMA GEMM:  H[M x N] = (relu?)(A[M x K]) @ W[K x N] ----
// One wave32 per 16x16 output tile. Block = (N/16) waves covering all N cols
// of a 16-row slab; grid.x = M/16 (M = 50000 is divisible by 16).
//
// V_WMMA_F32_16X16X4_F32 operand layout (ISA 7.12.2, wave32):
//   A (16x4):  lanes 0-15 -> M=lane, VGPR0=K0,VGPR1=K1; lanes 16-31 -> K2,K3
//   B (4x16):  lanes 0-15 -> N=lane, rows K0/K1;        lanes 16-31 -> K2/K3
//   C/D(16x16): VGPR r: lanes 0-15 -> M=r, N=lane; lanes 16-31 -> M=8+r
template<int K, int N, bool RELU_A>
__global__ void gemm16_wmma_f32(const float* __restrict__ A,
                                const float* __restrict__ W,
                                float* __restrict__ H) {
    const int lane  = threadIdx.x & 31;
    const int wave  = threadIdx.x >> 5;        // which 16-col tile
    const int row0  = blockIdx.x << 4;
    const int col0  = wave << 4;
    const int m     = lane & 15;               // M for A, N for B/C/D
    const int khalf = (lane >> 4) << 1;        // 0 for lanes 0-15, 2 for 16-31

    const float* __restrict__ arow = A + (size_t)(row0 + m) * K;
    const float* __restrict__ wcol = W + col0 + m;

    v8f acc = {};
#pragma unroll 8
    for (int k0 = 0; k0 < K; k0 += 4) {
        float a0 = arow[k0 + khalf + 0];
        float a1 = arow[k0 + khalf + 1];
        if (RELU_A) { a0 = fmaxf(a0, 0.0f); a1 = fmaxf(a1, 0.0f); }
        v2f a = {a0, a1};
        v2f b = { wcol[(size_t)(k0 + khalf + 0) * N],
                  wcol[(size_t)(k0 + khalf + 1) * N] };
        // (neg_a, A, neg_b, B, c_mod, C, reuse_a, reuse_b)
        acc = __builtin_amdgcn_wmma_f32_16x16x4_f32(
                  false, a, false, b, (short)0, acc, false, false);
    }

    float* __restrict__ hrow =
        H + (size_t)(row0 + ((lane >> 4) << 3)) * N + col0 + m;
#pragma unroll
    for (int r = 0; r < 8; ++r)
        hrow[(size_t)r * N] = acc[r];
}

// ---------------- self-loop + bias init:  out = h * isq[node]^2 + bias[c] ----
template<int C>
__global__ void selfloop_bias_kernel(const float* __restrict__ h,
                                     const float* __restrict__ isq,
                                     const float* __restrict__ bias,
                                     float* __restrict__ out, int n_elems) {
    int i = blockIdx.x * blockDim.x + threadIdx.x;
    if (i < n_elems) {
        int node = i / C;                      // C is power of two -> shift
        int c    = i - node * C;
        float s  = isq[node];
        out[i] = h[i] * (s * s) + bias[c];
    }
}

// ---------------- edge scatter-add:  out[dst] += h[src] * isq[s]*isq[d] -----
// One wave32 per edge (grid-stride). Lanes cover channels stride-1 so each
// gather / atomic burst is 128B contiguous per 32 lanes.
template<int C>
__global__ void edge_scatter_kernel(const float* __restrict__ h,
                                    const int* __restrict__ src,
                                    const int* __restrict__ dst,
                                    const float* __restrict__ isq,
                                    float* __restrict__ out, int E) {
    const int lane   = threadIdx.x & 31;
    const int wave   = (blockIdx.x * blockDim.x + threadIdx.x) >> 5;
    const int nwaves = (gridDim.x * blockDim.x) >> 5;
    constexpr int V  = C / 32;                 // floats per lane

    for (int e = wave; e < E; e += nwaves) {
        const int   s    = src[e];
        const int   d    = dst[e];
        const float coef = isq[s] * isq[d];
        const float* __restrict__ hs = h   + (size_t)s * C;
        float*       __restrict__ od = out + (size_t)d * C;
#pragma unroll
        for (int v = 0; v < V; ++v) {
            int c = lane + (v << 5);
            atomicAdd(&od[c], hs[c] * coef);
        }
    }
}

// ---------------------------------------------------------------------------

extern "C" void kernel_launch(void* const* d_in, const int* in_sizes, int n_in,
                              void* d_out, int out_size, void* d_ws, size_t ws_size,
                              hipStream_t stream) {
    (void)in_sizes; (void)n_in; (void)out_size; (void)ws_size;

    const float* x   = (const float*)d_in[0];          // [N, 128]
    const int*   ei  = (const int*)  d_in[1];          // [2, E] int32
    const float* W1  = (const float*)d_in[2];          // [128, 128]
    const float* b1  = (const float*)d_in[3];          // [128]
    const float* W2  = (const float*)d_in[4];          // [128, 64]
    const float* b2  = (const float*)d_in[5];          // [64]
    float*       out = (float*)d_out;                  // [N, 64]

    const int* src = ei;                               // row 0
    const int* dst = ei + N_EDGES;                     // row 1

    // workspace layout (floats): isq | h1 | out1 ; h2 aliases h1 (dead by then)
    float* isq  = (float*)d_ws;                        // 50176 (padded)
    float* h1   = isq + 50176;                         // N * 128
    float* out1 = h1 + (size_t)N_NODES * C_HID;        // N * 128
    float* h2   = h1;                                  // N * 64 (alias)

    const int T = 256;

    // 1) degree -> 1/sqrt(deg+1)
    zero_kernel   <<<(N_NODES + T - 1) / T, T, 0, stream>>>(isq, N_NODES);
    deg_kernel    <<<(N_EDGES + T - 1) / T, T, 0, stream>>>(dst, isq, N_EDGES);
    invsqrt_kernel<<<(N_NODES + T - 1) / T, T, 0, stream>>>(isq, N_NODES);

    // 2) layer 1: h1 = x @ W1   (8 waves/block, one per 16-col tile)
    gemm16_wmma_f32<C_IN, C_HID, false>
        <<<N_NODES / 16, (C_HID / 16) * 32, 0, stream>>>(x, W1, h1);

    // 3) out1 = h1 * isq^2 + b1 ; then scatter-add over edges
    selfloop_bias_kernel<C_HID>
        <<<((N_NODES * C_HID) + T - 1) / T, T, 0, stream>>>(h1, isq, b1, out1,
                                                            N_NODES * C_HID);
    edge_scatter_kernel<C_HID>
        <<<1024, T, 0, stream>>>(h1, src, dst, isq, out1, N_EDGES);

    // 4) layer 2: h2 = relu(out1) @ W2  (ReLU fused into A-load; 4 waves/block)
    gemm16_wmma_f32<C_HID, C_OUTC, true>
        <<<N_NODES / 16, (C_OUTC / 16) * 32, 0, stream>>>(out1, W2, h2);

    // 5) out = h2 * isq^2 + b2 ; then scatter-add over edges
    selfloop_bias_kernel<C_OUTC>
        <<<((N_NODES * C_OUTC) + T - 1) / T, T, 0, stream>>>(h2, isq, b2, out,
                                                             N_NODES * C_OUTC);
    edge_scatter_kernel<C_OUTC>
        <<<1024, T, 0, stream>>>(h2, src, dst, isq, out, N_EDGES);
}